// RecurTreeGen_35270271434818
// MI455X (gfx1250) — compile-verified
//
#include <hip/hip_runtime.h>
#include <math.h>

typedef __attribute__((ext_vector_type(2))) float v2f;
typedef __attribute__((ext_vector_type(4))) float v4f;
typedef __attribute__((ext_vector_type(8))) float v8f;

#define TDIM 128      // hidden dim
#define TK   256      // 2*DIM (GEMM K)
#define TGN  640      // 5*DIM (GEMM N)
#define XLD  260      // padded LDS row stride for x: (4*l15 + k) % 64 unique across a wave
#define MROWS 32      // nodes per block (two 16-row WMMA M-tiles, B-fragments reused 2x)

__device__ __forceinline__ float rcp_fast(float x) { return __builtin_amdgcn_rcpf(x); }
__device__ __forceinline__ float sigf(float x) {
    return rcp_fast(1.0f + __expf(-x));
}
// tanh(x) = 1 - 2/(exp(2x)+1): saturates cleanly at +/-1, no inf/inf NaN
__device__ __forceinline__ float tanh_fast(float x) {
    return 1.0f - 2.0f * rcp_fast(__expf(2.0f * x) + 1.0f);
}

__global__ __launch_bounds__(256)
void tree_lstm_level_kernel(const float* __restrict__ hp, const float* __restrict__ cp,
                            const float* __restrict__ W,  const float* __restrict__ bias,
                            const int* __restrict__ li,   const int* __restrict__ ri,
                            float* __restrict__ hout, float* __restrict__ cout, int n)
{
    __shared__ float xs [MROWS * XLD];    // concat(h_l, h_r): 32 x 256, padded rows
    __shared__ float cls[MROWS * TDIM];   // c_l
    __shared__ float crs[MROWS * TDIM];   // c_r

    const int tid   = threadIdx.x;
    const int lane  = tid & 31;
    const int wave  = tid >> 5;        // 0..7
    const int l15   = lane & 15;
    const int half  = lane >> 4;       // 0: lanes 0-15, 1: lanes 16-31
    const int node0 = blockIdx.x * MROWS;

    // ---- gather: wave w loads rows 4w..4w+3; 32 lanes x float4 == 128 floats/row ----
    #pragma unroll
    for (int rr = 0; rr < 4; ++rr) {
        const int m    = wave * 4 + rr;
        const int node = node0 + m;
        const int gl = (node < n) ? li[node] : 0;
        const int gr = (node < n) ? ri[node] : 0;
        v4f vhl = *(const v4f*)(hp + (size_t)gl * TDIM + lane * 4);
        v4f vhr = *(const v4f*)(hp + (size_t)gr * TDIM + lane * 4);
        v4f vcl = *(const v4f*)(cp + (size_t)gl * TDIM + lane * 4);
        v4f vcr = *(const v4f*)(cp + (size_t)gr * TDIM + lane * 4);
        *(v4f*)(&xs [m * XLD  + lane * 4])        = vhl;
        *(v4f*)(&xs [m * XLD  + TDIM + lane * 4]) = vhr;
        *(v4f*)(&cls[m * TDIM + lane * 4])        = vcl;
        *(v4f*)(&crs[m * TDIM + lane * 4])        = vcr;
    }
    __syncthreads();

    // Wave w owns N-tiles {g*8 + w} == gate g, output dims [16w, 16w+16),
    // for BOTH 16-row M-tiles (mt=0: rows 0-15, mt=1: rows 16-31).
    const int dcol = wave * 16 + l15;  // 0..127, this lane's output dim

    v8f acc[5][2];
    #pragma unroll
    for (int g = 0; g < 5; ++g) {
        const float bb = bias[g * TDIM + dcol]; // per-column bias, same for all rows
        #pragma unroll
        for (int r = 0; r < 8; ++r) { acc[g][0][r] = bb; acc[g][1][r] = bb; }
    }

    const float* wcol = W + dcol;      // column base for this lane

    // ---- GEMM: [32 x 256] @ [256 x 640], fp32 WMMA 16x16x4; each B fragment
    //      is loaded once and fed to two WMMAs (two M-tiles) ----
    #pragma unroll 2
    for (int kk = 0; kk < TK; kk += 4) {
        const int ka = kk + half * 2;  // ISA layout: lanes 16-31 hold K=2,3
        v2f a0 = *(const v2f*)(&xs[(l15     ) * XLD + ka]);  // rows 0-15
        v2f a1 = *(const v2f*)(&xs[(l15 + 16) * XLD + ka]);  // rows 16-31
        #pragma unroll
        for (int g = 0; g < 5; ++g) {
            v2f bf;
            bf.x = wcol[(size_t)(ka + 0) * TGN + g * TDIM];
            bf.y = wcol[(size_t)(ka + 1) * TGN + g * TDIM];
            acc[g][0] = __builtin_amdgcn_wmma_f32_16x16x4_f32(
                false, a0, false, bf, (short)0, acc[g][0], false, false);
            acc[g][1] = __builtin_amdgcn_wmma_f32_16x16x4_f32(
                false, a1, false, bf, (short)0, acc[g][1], false, false);
        }
    }

    // ---- LSTM cell entirely in registers: acc[0..4] = i,o,u,fl,fr for dim dcol ----
    // C/D layout: VGPR r -> row r (lanes 0-15) or row r+8 (lanes 16-31)
    #pragma unroll
    for (int mt = 0; mt < 2; ++mt) {
        #pragma unroll
        for (int r = 0; r < 8; ++r) {
            const int m    = mt * 16 + r + half * 8;
            const int node = node0 + m;
            const float gi  = acc[0][mt][r];
            const float go  = acc[1][mt][r];
            const float gu  = acc[2][mt][r];
            const float gfl = acc[3][mt][r];
            const float gfr = acc[4][mt][r];
            const float cva = sigf(gi)  * tanh_fast(gu)
                            + sigf(gfl) * cls[m * TDIM + dcol]
                            + sigf(gfr) * crs[m * TDIM + dcol];
            const float hva = sigf(go) * tanh_fast(cva);
            if (node < n) {
                hout[(size_t)node * TDIM + dcol] = hva;
                cout[(size_t)node * TDIM + dcol] = cva;
            }
        }
    }
}

extern "C" void kernel_launch(void* const* d_in, const int* in_sizes, int n_in,
                              void* d_out, int out_size, void* d_ws, size_t ws_size,
                              hipStream_t stream)
{
    const float* h_leaf = (const float*)d_in[0];
    const float* c_leaf = (const float*)d_in[1];
    const float* W      = (const float*)d_in[2];
    const float* bias   = (const float*)d_in[3];
    const int*   li     = (const int*)d_in[4];
    const int*   ri     = (const int*)d_in[5];

    float* out   = (float*)d_out;
    float* c_all = (float*)d_ws;  // needs (N_LEAF-1)*128*4 bytes (~134 MB) of scratch

    const int n_leaf = in_sizes[0] / TDIM;  // 262144

    // d_out doubles as h_prev storage for the next level (it is exactly the
    // concatenated per-level h); c goes to workspace at matching offsets.
    const float* hp = h_leaf;
    const float* cp = c_leaf;
    size_t off = 0;
    for (int nn = n_leaf / 2; nn >= 1; nn >>= 1) {
        const int blocks = (nn + MROWS - 1) / MROWS;
        tree_lstm_level_kernel<<<blocks, 256, 0, stream>>>(
            hp, cp, W, bias, li + off, ri + off,
            out + off * TDIM, c_all + off * TDIM, nn);
        hp = out   + off * TDIM;   // next level gathers from this level's h
        cp = c_all + off * TDIM;   // and this level's c
        off += nn;
    }
}